// Cy2C_GCN_attr_1_30039001268370
// MI455X (gfx1250) — compile-verified
//
#include <hip/hip_runtime.h>

#define N_NODES 50000
#define N_GRAPHS 512
#define HID 128
#define E_EDGES 800000
#define E_CYC 400000
#define BN_EPS 1e-5f

typedef __attribute__((ext_vector_type(2))) float v2f;
typedef __attribute__((ext_vector_type(8))) float v8f;

// ---------------- WMMA GEMM: Y[M,128] = X[M,K] @ W[K,128] (+ bias) ----------------
// Block = 256 threads = 8 waves. Block handles 16 rows; wave w handles 16-col tile w.
// A tile (16 x K) staged in LDS via gfx1250 async-to-LDS DMA, shared by all 8 waves.
// Optional fused GCN epilogue: agg[r,c] = acc[r,c]/deg[r] + bias2[c] (self-loop+bias
// init for the scatter phase) written alongside the raw xw output.
__global__ void gemm_wmma_n128(const float* __restrict__ X, const float* __restrict__ W,
                               const float* __restrict__ bias, float* __restrict__ Y, int K,
                               const float* __restrict__ deg, const float* __restrict__ bias2,
                               float* __restrict__ agg_out) {
    __shared__ float sX[16 * 256];              // supports K up to 256
    const int tid  = threadIdx.x;
    const int wave = tid >> 5;
    const int lane = tid & 31;
    const int row0 = blockIdx.x * 16;
    const int col0 = wave * 16;

    // 16 tile rows are contiguous in X (row-major): async DMA straight into LDS.
    // bytes = 16*K*4 (8KB for K=128, 16KB for K=256) — exact multiple of 256*16B,
    // so the loop is uniform across the wave (EXEC all-ones per iteration).
    {
        const float* srcp = X + (size_t)row0 * K;           // uniform (SGPR pair)
        const unsigned bytes   = (unsigned)(16 * K * 4);
        const unsigned ldsbase = (unsigned)(size_t)(&sX[0]);
        for (unsigned off = (unsigned)tid * 16u; off < bytes; off += 256u * 16u) {
            asm volatile("global_load_async_to_lds_b128 %0, %1, %2"
                         :: "v"(ldsbase + off), "v"(off), "s"(srcp)
                         : "memory");
        }
        asm volatile("s_wait_asynccnt 0x0" ::: "memory");
    }
    __syncthreads();

    const int m    = lane & 15;                 // A: M index / B,C,D: N index
    const int half = lane >> 4;                 // K-pair / M-half selector

    v8f acc = {};
#pragma unroll 8
    for (int k0 = 0; k0 < K; k0 += 4) {
        const int ka = k0 + half * 2;
        v2f a, b;
        a.x = sX[m * K + ka + 0];
        a.y = sX[m * K + ka + 1];
        b.x = W[(ka + 0) * HID + col0 + m];
        b.y = W[(ka + 1) * HID + col0 + m];
        acc = __builtin_amdgcn_wmma_f32_16x16x4_f32(false, a, false, b,
                                                    (short)0, acc, false, false);
    }

    const float bv  = bias  ? bias[col0 + m]  : 0.0f;
    const float bv2 = bias2 ? bias2[col0 + m] : 0.0f;
#pragma unroll
    for (int v = 0; v < 8; ++v) {
        const int row = row0 + v + half * 8;    // C/D layout: M = v + (lane>=16)*8
        const size_t idx = (size_t)row * HID + col0 + m;
        Y[idx] = acc[v] + bv;
        if (agg_out) agg_out[idx] = acc[v] / deg[row] + bv2;
    }
}

// ---------------- degree = 1 (self loop) + segment_sum(edge_weight by dst) --------
__global__ void init_deg(float* d1, float* d2, int n) {
    int i = blockIdx.x * blockDim.x + threadIdx.x;
    if (i < n) { d1[i] = 1.0f; d2[i] = 1.0f; }
}

__global__ void accum_deg(const int* __restrict__ dst, const float* __restrict__ ew,
                          float* __restrict__ deg, int E) {
    int e = blockIdx.x * blockDim.x + threadIdx.x;
    if (e < E) atomicAdd(&deg[dst[e]], ew ? ew[e] : 1.0f);
}

// ---------------- edge scatter: one wave per edge, 4 cols per lane ----------------
__global__ void scatter_edges(const int* __restrict__ src, const int* __restrict__ dst,
                              const float* __restrict__ ew, const float* __restrict__ deg,
                              const float* __restrict__ xw, float* __restrict__ agg, int E) {
    int g = blockIdx.x * blockDim.x + threadIdx.x;
    int e = g >> 5, lane = g & 31;
    if (e >= E) return;
    const int s = src[e], d = dst[e];
    const float w = ew ? ew[e] : 1.0f;
    const float norm = rsqrtf(deg[s]) * w * rsqrtf(deg[d]);
    const float4 v = *(const float4*)(xw + (size_t)s * HID + lane * 4);
    float* base = agg + (size_t)d * HID + lane * 4;
    atomicAdd(base + 0, v.x * norm);
    atomicAdd(base + 1, v.y * norm);
    atomicAdd(base + 2, v.z * norm);
    atomicAdd(base + 3, v.w * norm);
}

// ---------------- batchnorm: column sum / sumsq, then scale/shift -----------------
__global__ void col_stats(const float* __restrict__ y, float* __restrict__ sum,
                          float* __restrict__ sumsq, int N) {
    const int tid = threadIdx.x;
    const int c = tid & 127;
    int r = blockIdx.x * 256 + (tid >> 7);
    int rend = blockIdx.x * 256 + 256; if (rend > N) rend = N;
    float s = 0.0f, ss = 0.0f;
    for (; r < rend; r += 2) {
        float v = y[(size_t)r * HID + c];
        s += v; ss += v * v;
    }
    atomicAdd(&sum[c], s);
    atomicAdd(&sumsq[c], ss);
}

__global__ void finalize_stats(const float* __restrict__ sum, const float* __restrict__ sumsq,
                               const float* __restrict__ g, const float* __restrict__ b,
                               float* __restrict__ scale, float* __restrict__ shift, float invN) {
    int c = threadIdx.x;
    float m = sum[c] * invN;
    float var = sumsq[c] * invN - m * m;      // biased var, matches jnp.var
    float sc = g[c] * rsqrtf(var + BN_EPS);
    scale[c] = sc;
    shift[c] = b[c] - m * sc;
}

__global__ void bn_relu_res(const float* __restrict__ y, const float* __restrict__ scale,
                            const float* __restrict__ shift, const float* __restrict__ res,
                            float* __restrict__ out, int total) {
    int i = blockIdx.x * blockDim.x + threadIdx.x;
    if (i < total) {
        int c = i & 127;
        float v = y[i] * scale[c] + shift[c];
        v = v > 0.0f ? v : 0.0f;
        out[i] = v + res[i];
    }
}

// ---------------- mean pool ------------------------------------------------------
__global__ void zero_buf(float* p, int n) {
    int i = blockIdx.x * blockDim.x + threadIdx.x;
    if (i < n) p[i] = 0.0f;
}

__global__ void pool_accum(const float* __restrict__ x, const int* __restrict__ batch,
                           float* __restrict__ pooled, float* __restrict__ counts,
                           int col_off, int N) {
    int g = blockIdx.x * blockDim.x + threadIdx.x;
    int n = g >> 5, lane = g & 31;
    if (n >= N) return;
    const int bb = batch[n];
    const float4 v = *(const float4*)(x + (size_t)n * HID + lane * 4);
    float* base = pooled + (size_t)bb * (2 * HID) + col_off + lane * 4;
    atomicAdd(base + 0, v.x);
    atomicAdd(base + 1, v.y);
    atomicAdd(base + 2, v.z);
    atomicAdd(base + 3, v.w);
    if (lane == 0 && col_off == 0) atomicAdd(&counts[bb], 1.0f);
}

__global__ void pool_div(float* __restrict__ pooled, const float* __restrict__ counts, int total) {
    int i = blockIdx.x * blockDim.x + threadIdx.x;
    if (i < total) pooled[i] /= fmaxf(counts[i >> 8], 1.0f);
}

// ---------------------------------------------------------------------------------
extern "C" void kernel_launch(void* const* d_in, const int* in_sizes, int n_in,
                              void* d_out, int out_size, void* d_ws, size_t ws_size,
                              hipStream_t stream) {
    const float* x          = (const float*)d_in[0];
    const int*   edge_index = (const int*)  d_in[1];
    const int*   cycle_index= (const int*)  d_in[2];
    const int*   batch      = (const int*)  d_in[3];
    const float* edge_attr  = (const float*)d_in[4];
    const float* emb_W      = (const float*)d_in[5];
    const float* emb_b      = (const float*)d_in[6];
    const float* conv_W     = (const float*)d_in[7];
    const float* conv_b     = (const float*)d_in[8];
    const float* bn_g       = (const float*)d_in[9];
    const float* bn_b       = (const float*)d_in[10];
    const float* conv2_W    = (const float*)d_in[11];
    const float* conv2_b    = (const float*)d_in[12];
    const float* bn2_g      = (const float*)d_in[13];
    const float* bn2_b      = (const float*)d_in[14];
    const float* lin1_W     = (const float*)d_in[15];
    const float* lin1_b     = (const float*)d_in[16];
    float* out = (float*)d_out;

    float* ws = (float*)d_ws;
    float* x0     = ws; ws += (size_t)N_NODES * HID;
    float* h      = ws; ws += (size_t)N_NODES * HID;
    float* xw     = ws; ws += (size_t)N_NODES * HID;
    float* agg    = ws; ws += (size_t)N_NODES * HID;
    float* deg_e  = ws; ws += N_NODES;
    float* deg_c  = ws; ws += N_NODES;
    float* pooled = ws; ws += N_GRAPHS * 2 * HID;   // counts follows contiguously
    float* counts = ws; ws += N_GRAPHS;
    float* ssum   = ws; ws += HID;                  // ssq follows contiguously
    float* ssq    = ws; ws += HID;
    float* scale  = ws; ws += HID;
    float* shift  = ws; ws += HID;

    const int* e_src = edge_index;
    const int* e_dst = edge_index + E_EDGES;
    const int* c_src = cycle_index;
    const int* c_dst = cycle_index + E_CYC;

    const int TOT = N_NODES * HID;

    // degrees (self-loop weight 1 baked into init)
    init_deg<<<(N_NODES + 255) / 256, 256, 0, stream>>>(deg_e, deg_c, N_NODES);
    accum_deg<<<(E_EDGES + 255) / 256, 256, 0, stream>>>(e_dst, edge_attr, deg_e, E_EDGES);
    accum_deg<<<(E_CYC + 255) / 256, 256, 0, stream>>>(c_dst, nullptr, deg_c, E_CYC);

    // x0 = x @ emb_W + emb_b ; h = x0
    gemm_wmma_n128<<<N_NODES / 16, 256, 0, stream>>>(x, emb_W, emb_b, x0, HID,
                                                     nullptr, nullptr, nullptr);
    hipMemcpyAsync(h, x0, (size_t)TOT * sizeof(float), hipMemcpyDeviceToDevice, stream);

    // three residual GCN layers on edge_index (agg init fused into GEMM epilogue)
    for (int i = 0; i < 3; ++i) {
        gemm_wmma_n128<<<N_NODES / 16, 256, 0, stream>>>(h, conv_W + (size_t)i * HID * HID,
                                                         nullptr, xw, HID,
                                                         deg_e, conv_b + i * HID, agg);
        scatter_edges<<<(E_EDGES * 32 + 255) / 256, 256, 0, stream>>>(e_src, e_dst, edge_attr,
                                                                      deg_e, xw, agg, E_EDGES);
        zero_buf<<<1, 256, 0, stream>>>(ssum, 2 * HID);   // ssum + ssq
        col_stats<<<(N_NODES + 255) / 256, 256, 0, stream>>>(agg, ssum, ssq, N_NODES);
        finalize_stats<<<1, 128, 0, stream>>>(ssum, ssq, bn_g + i * HID, bn_b + i * HID,
                                              scale, shift, 1.0f / N_NODES);
        bn_relu_res<<<(TOT + 255) / 256, 256, 0, stream>>>(agg, scale, shift, h, h, TOT);
    }

    // cycle branch: xb = relu(bn(gcn(x0, cycle_index, ones))) + x0  -> stored in agg
    gemm_wmma_n128<<<N_NODES / 16, 256, 0, stream>>>(x0, conv2_W, nullptr, xw, HID,
                                                     deg_c, conv2_b, agg);
    scatter_edges<<<(E_CYC * 32 + 255) / 256, 256, 0, stream>>>(c_src, c_dst, nullptr,
                                                                deg_c, xw, agg, E_CYC);
    zero_buf<<<1, 256, 0, stream>>>(ssum, 2 * HID);
    col_stats<<<(N_NODES + 255) / 256, 256, 0, stream>>>(agg, ssum, ssq, N_NODES);
    finalize_stats<<<1, 128, 0, stream>>>(ssum, ssq, bn2_g, bn2_b, scale, shift, 1.0f / N_NODES);
    bn_relu_res<<<(TOT + 255) / 256, 256, 0, stream>>>(agg, scale, shift, x0, agg, TOT);

    // mean pool both branches into concat buffer [512, 256]
    zero_buf<<<(N_GRAPHS * 2 * HID + N_GRAPHS + 255) / 256, 256, 0, stream>>>(
        pooled, N_GRAPHS * 2 * HID + N_GRAPHS);   // pooled + counts
    pool_accum<<<(N_NODES * 32 + 255) / 256, 256, 0, stream>>>(h,   batch, pooled, counts, 0,   N_NODES);
    pool_accum<<<(N_NODES * 32 + 255) / 256, 256, 0, stream>>>(agg, batch, pooled, counts, HID, N_NODES);
    pool_div<<<(N_GRAPHS * 2 * HID + 255) / 256, 256, 0, stream>>>(pooled, counts,
                                                                   N_GRAPHS * 2 * HID);

    // out[512,128] = pooled[512,256] @ lin1_W + lin1_b
    gemm_wmma_n128<<<N_GRAPHS / 16, 256, 0, stream>>>(pooled, lin1_W, lin1_b, out, 2 * HID,
                                                      nullptr, nullptr, nullptr);
}